// GCN_single_84971632984515
// MI455X (gfx1250) — compile-verified
//
#include <hip/hip_runtime.h>
#include <math.h>

typedef float v2f __attribute__((ext_vector_type(2)));
typedef float v8f __attribute__((ext_vector_type(8)));

#define FEAT 64

// ---------------- elementwise / setup kernels ----------------

__global__ void fill1_kernel(float* __restrict__ p, int n) {
    int i = blockIdx.x * blockDim.x + threadIdx.x;
    int s = gridDim.x * blockDim.x;
    for (; i < n; i += s) p[i] = 1.0f;
}

// deg layout: slot t-1 (t=1..4) = in-degree of type t ; slot t+2 (t=2..4) = out-degree of type t
__global__ void degree_kernel(const int* __restrict__ esrc, const int* __restrict__ edst,
                              const int* __restrict__ etype, float* __restrict__ deg,
                              int nE, int nN) {
    int i = blockIdx.x * blockDim.x + threadIdx.x;
    int s = gridDim.x * blockDim.x;
    for (int e = i; e < nE; e += s) {
        int t = etype[e];
        if (t >= 1) atomicAdd(deg + (size_t)(t - 1) * nN + edst[e], 1.0f);
        if (t >= 2) atomicAdd(deg + (size_t)(t + 2) * nN + esrc[e], 1.0f);
    }
}

__global__ void rsqrt_kernel(float* __restrict__ p, int n) {
    int i = blockIdx.x * blockDim.x + threadIdx.x;
    int s = gridDim.x * blockDim.x;
    for (; i < n; i += s) p[i] = rsqrtf(p[i]);
}

__global__ void bsum_kernel(const float* b0, const float* b1, const float* b2,
                            const float* b3, const float* b4, const float* b5,
                            const float* b6, float* __restrict__ bs) {
    int d = threadIdx.x;
    bs[d] = b0[d] + b1[d] + b2[d] + b3[d] + b4[d] + b5[d] + b6[d];
}

__global__ void h0_kernel(const float* __restrict__ x, const float* __restrict__ nW,
                          const float* __restrict__ nb, float* __restrict__ h, int total) {
    int i = blockIdx.x * blockDim.x + threadIdx.x;
    int s = gridDim.x * blockDim.x;
    for (; i < total; i += s) {
        int n = i >> 6, d = i & 63;
        h[i] = fmaf(x[n], nW[d], nb[d]);
    }
}

__global__ void init_hnew_kernel(float4* __restrict__ hn, const float4* __restrict__ bs, int n4) {
    int i = blockIdx.x * blockDim.x + threadIdx.x;
    int s = gridDim.x * blockDim.x;
    for (; i < n4; i += s) hn[i] = bs[i & 15];
}

// A = dinv^2 * h  (self-loop term folded into scatter target)
__global__ void initA_kernel(float4* __restrict__ A, const float4* __restrict__ h,
                             const float* __restrict__ dinv, int n4) {
    int i = blockIdx.x * blockDim.x + threadIdx.x;
    int s = gridDim.x * blockDim.x;
    for (; i < n4; i += s) {
        int n = i >> 4;                 // 16 float4 per 64-wide row
        float dv = dinv[n];
        float sc = dv * dv;
        float4 v = h[i];
        v.x *= sc; v.y *= sc; v.z *= sc; v.w *= sc;
        A[i] = v;
    }
}

__global__ void relu_kernel(float4* __restrict__ h, const float4* __restrict__ hn, int n4) {
    int i = blockIdx.x * blockDim.x + threadIdx.x;
    int s = gridDim.x * blockDim.x;
    for (; i < n4; i += s) {
        float4 v = hn[i];
        v.x = fmaxf(v.x, 0.f); v.y = fmaxf(v.y, 0.f);
        v.z = fmaxf(v.z, 0.f); v.w = fmaxf(v.w, 0.f);
        h[i] = v;
    }
}

// ---------------- edge scatter: one wave per edge, lane handles 2 feature floats ----------------

__global__ void scatter_kernel(const int* __restrict__ esrc, const int* __restrict__ edst,
                               const int* __restrict__ etype, const float* __restrict__ dinv,
                               const float* __restrict__ h, float* __restrict__ A,
                               int ktype, int reversed, int nE) {
    int lane  = threadIdx.x & 31;
    int wave  = (blockIdx.x * blockDim.x + threadIdx.x) >> 5;
    int nWave = (gridDim.x * blockDim.x) >> 5;
    for (int e = wave; e < nE; e += nWave) {
        if (etype[e] != ktype) continue;
        int s = esrc[e], d = edst[e];
        int gn = reversed ? d : s;      // gather node
        int sn = reversed ? s : d;      // scatter node
        float nrm = dinv[gn] * dinv[sn];
        const float2 v = *(const float2*)(h + (size_t)gn * FEAT + 2 * lane);
        float* ap = A + (size_t)sn * FEAT + 2 * lane;
        atomicAdd(ap,     v.x * nrm);
        atomicAdd(ap + 1, v.y * nrm);
    }
}

// ---------------- WMMA f32 GEMM-accumulate: hnew[16x64 tile] += A[16x64] @ W[64x64] ----------------

__global__ void __launch_bounds__(256) gemm_acc_kernel(const float* __restrict__ A,
                                                       const float* __restrict__ W,
                                                       float* __restrict__ Hn) {
    __shared__ float sW[FEAT * FEAT];
    // stage W (16 KB) into LDS
    for (int i = threadIdx.x; i < (FEAT * FEAT) / 4; i += 256)
        ((float4*)sW)[i] = ((const float4*)W)[i];
    __syncthreads();

    int wave = threadIdx.x >> 5;
    int lane = threadIdx.x & 31;
    int hf   = lane >> 4;               // half-wave select
    int lr   = lane & 15;               // M (A rows) / N (B,C cols) index
    int rowBase = blockIdx.x * 128 + wave * 16;

    // C layout (ISA 7.12.2): VGPR j, lanes0-15 -> M=j ; lanes16-31 -> M=j+8 ; N = lr
    v8f c0, c1, c2, c3;
    float* hp = Hn + (size_t)rowBase * FEAT;
#pragma unroll
    for (int j = 0; j < 8; ++j) {
        int r = (j + 8 * hf) * FEAT + lr;
        c0[j] = hp[r];
        c1[j] = hp[r + 16];
        c2[j] = hp[r + 32];
        c3[j] = hp[r + 48];
    }

    const float* ap = A + (size_t)(rowBase + lr) * FEAT;
#pragma unroll
    for (int k0 = 0; k0 < FEAT; k0 += 4) {
        // A frag (16x4 f32): lanes0-15 K={k0,k0+1}, lanes16-31 K={k0+2,k0+3}, M=lr
        v2f a = *(const v2f*)(ap + k0 + 2 * hf);
        int krow = (k0 + 2 * hf) * FEAT;
        v2f b0, b1, b2, b3;
        b0.x = sW[krow + lr];            b0.y = sW[krow + FEAT + lr];
        b1.x = sW[krow + 16 + lr];       b1.y = sW[krow + FEAT + 16 + lr];
        b2.x = sW[krow + 32 + lr];       b2.y = sW[krow + FEAT + 32 + lr];
        b3.x = sW[krow + 48 + lr];       b3.y = sW[krow + FEAT + 48 + lr];
        c0 = __builtin_amdgcn_wmma_f32_16x16x4_f32(false, a, false, b0, (short)0, c0, false, false);
        c1 = __builtin_amdgcn_wmma_f32_16x16x4_f32(false, a, false, b1, (short)0, c1, false, false);
        c2 = __builtin_amdgcn_wmma_f32_16x16x4_f32(false, a, false, b2, (short)0, c2, false, false);
        c3 = __builtin_amdgcn_wmma_f32_16x16x4_f32(false, a, false, b3, (short)0, c3, false, false);
    }

#pragma unroll
    for (int j = 0; j < 8; ++j) {
        int r = (j + 8 * hf) * FEAT + lr;
        hp[r]      = c0[j];
        hp[r + 16] = c1[j];
        hp[r + 32] = c2[j];
        hp[r + 48] = c3[j];
    }
}

// ---------------- graph max-pool + 2-layer MLP head: one wave per graph ----------------

__global__ void __launch_bounds__(256) pool_mlp_kernel(const float* __restrict__ h,
                                                       const float* __restrict__ W1o,
                                                       const float* __restrict__ b1o,
                                                       const float* __restrict__ W2o,
                                                       const float* __restrict__ b2o,
                                                       float* __restrict__ out, int nG) {
    __shared__ float sg[8 * FEAT];
    int wave = threadIdx.x >> 5;
    int lane = threadIdx.x & 31;
    int b = blockIdx.x * 8 + wave;
    if (b >= nG) return;
    const float* hb = h + (size_t)b * 64 * FEAT;
    float m0 = -3.402823466e38f, m1 = -3.402823466e38f;
#pragma unroll 4
    for (int i = 0; i < 64; ++i) {
        float2 v = *(const float2*)(hb + i * FEAT + 2 * lane);
        m0 = fmaxf(m0, v.x);
        m1 = fmaxf(m1, v.y);
    }
    sg[wave * FEAT + 2 * lane]     = m0;
    sg[wave * FEAT + 2 * lane + 1] = m1;
    __syncthreads();
    int d0 = 2 * lane, d1 = 2 * lane + 1;
    float a0 = b1o[d0], a1 = b1o[d1];
    const float* g = sg + wave * FEAT;
#pragma unroll 4
    for (int k = 0; k < FEAT; ++k) {
        float gk = g[k];
        a0 = fmaf(gk, W1o[k * FEAT + d0], a0);
        a1 = fmaf(gk, W1o[k * FEAT + d1], a1);
    }
    a0 = fmaxf(a0, 0.f);
    a1 = fmaxf(a1, 0.f);
    float r = a0 * W2o[d0] + a1 * W2o[d1];
#pragma unroll
    for (int off = 16; off > 0; off >>= 1) r += __shfl_down(r, off, 32);
    if (lane == 0) out[b] = r + b2o[0];
}

// ---------------- launcher ----------------

extern "C" void kernel_launch(void* const* d_in, const int* in_sizes, int n_in,
                              void* d_out, int out_size, void* d_ws, size_t ws_size,
                              hipStream_t stream) {
    const float* x     = (const float*)d_in[0];
    const int*   eidx  = (const int*)  d_in[1];   // [2, E]
    const int*   etype = (const int*)  d_in[2];
    const float* nodeW = (const float*)d_in[3];
    const float* nodeb = (const float*)d_in[4];
    const float* o1W   = (const float*)d_in[5];
    const float* o1b   = (const float*)d_in[6];
    const float* o2W   = (const float*)d_in[7];
    const float* o2b   = (const float*)d_in[8];
    const float* Wc[7];
    const float* bc[7];
    for (int c = 0; c < 7; ++c) {
        Wc[c] = (const float*)d_in[9 + 2 * c];
        bc[c] = (const float*)d_in[10 + 2 * c];
    }

    const int nN = in_sizes[0];     // 262144 nodes (x is [N,1])
    const int nE = in_sizes[2];     // 4194304 edges
    const int* esrc = eidx;
    const int* edst = eidx + nE;

    float* ws   = (float*)d_ws;
    float* h    = ws;                            // nN*64
    float* hn   = h  + (size_t)nN * FEAT;        // nN*64
    float* A    = hn + (size_t)nN * FEAT;        // nN*64
    float* deg  = A  + (size_t)nN * FEAT;        // 7*nN  (dinv after rsqrt)
    float* bsum = deg + (size_t)7 * nN;          // 64

    // conv table: forward types 1..4 use in-degree slots 0..3; reversed 2..4 use out-degree slots 4..6
    const int ktype[7] = {1, 2, 3, 4, 2, 3, 4};
    const int krev[7]  = {0, 0, 0, 0, 1, 1, 1};

    const int total = nN * FEAT;
    const int n4 = total / 4;

    fill1_kernel <<<2048, 256, 0, stream>>>(deg, 7 * nN);
    degree_kernel<<<2048, 256, 0, stream>>>(esrc, edst, etype, deg, nE, nN);
    rsqrt_kernel <<<2048, 256, 0, stream>>>(deg, 7 * nN);
    bsum_kernel  <<<1, 64, 0, stream>>>(bc[0], bc[1], bc[2], bc[3], bc[4], bc[5], bc[6], bsum);
    h0_kernel    <<<4096, 256, 0, stream>>>(x, nodeW, nodeb, h, total);

    for (int layer = 0; layer < 3; ++layer) {
        init_hnew_kernel<<<4096, 256, 0, stream>>>((float4*)hn, (const float4*)bsum, n4);
        for (int c = 0; c < 7; ++c) {
            const float* dv = deg + (size_t)c * nN;
            initA_kernel  <<<4096, 256, 0, stream>>>((float4*)A, (const float4*)h, dv, n4);
            scatter_kernel<<<4096, 256, 0, stream>>>(esrc, edst, etype, dv, h, A,
                                                     ktype[c], krev[c], nE);
            gemm_acc_kernel<<<nN / 128, 256, 0, stream>>>(A, Wc[c], hn);
        }
        relu_kernel<<<4096, 256, 0, stream>>>((float4*)h, (const float4*)hn, n4);
    }

    int nG = nN / 64;
    pool_mlp_kernel<<<(nG + 7) / 8, 256, 0, stream>>>(h, o1W, o1b, o2W, o2b,
                                                      (float*)d_out, nG);
}